// GCNEncoder_23854248362194
// MI455X (gfx1250) — compile-verified
//
#include <hip/hip_runtime.h>

// ---- problem constants (match reference setup_inputs) ----
#define NNODES 50000
#define NEDGES 800000
#define C_IN   128
#define C_HID  128
#define C_OUT  64

typedef __attribute__((ext_vector_type(2))) float v2f;
typedef __attribute__((ext_vector_type(8))) float v8f;

// ---------------- degree / dinv ----------------
__global__ void k_init_deg(float* __restrict__ deg) {
    int i = blockIdx.x * blockDim.x + threadIdx.x;
    if (i < NNODES) deg[i] = 1.0f;           // self-loop contributes 1 to every node
}

__global__ void k_deg_edges(const long long* __restrict__ dst, float* __restrict__ deg) {
    int e = blockIdx.x * blockDim.x + threadIdx.x;
    if (e < NEDGES) atomicAdd(&deg[(int)dst[e]], 1.0f);
}

__global__ void k_dinv(float* __restrict__ deg) {
    int i = blockIdx.x * blockDim.x + threadIdx.x;
    if (i < NNODES) deg[i] = rsqrtf(deg[i]); // deg >= 1 always (self-loops)
}

// ---------------- fp32 WMMA GEMM with fused row scale: G = (A @ W) * dinv[row] ----
// One wave computes a 16-row strip across all NCOL columns.
// A fragment (16x4 f32): lane<16 -> M=lane, K pair {k,k+1}; lane>=16 -> M=lane-16, K pair {k+2,k+3}
// B fragment (4x16 f32): mirrored (N=lane&15, K pair by lane half)
// C/D (16x16 f32): VGPR r holds M=r (lanes 0-15) / M=r+8 (lanes 16-31), N=lane&15
template <int K, int NCOL>
__global__ void k_gemm_scale(const float* __restrict__ A,
                             const float* __restrict__ W,
                             const float* __restrict__ dinv,
                             float* __restrict__ G) {
    constexpr int NT = NCOL / 16;
    const int wave = (blockIdx.x * blockDim.x + threadIdx.x) >> 5;
    const int lane = threadIdx.x & 31;
    const int m0   = wave * 16;
    if (m0 >= NNODES) return;                // wave-uniform: EXEC stays all-1s for WMMA

    const int l    = lane & 15;
    const int half = lane >> 4;
    const int kb   = half * 2;

    v8f acc[NT];
#pragma unroll
    for (int t = 0; t < NT; ++t) acc[t] = (v8f)0.0f;

    const float* __restrict__ arow = A + (size_t)(m0 + l) * K;

    for (int k = 0; k < K; k += 4) {
        v2f a;
        a.x = arow[k + kb];
        a.y = arow[k + kb + 1];
#pragma unroll
        for (int t = 0; t < NT; ++t) {
            v2f b;
            b.x = W[(size_t)(k + kb)     * NCOL + t * 16 + l];
            b.y = W[(size_t)(k + kb + 1) * NCOL + t * 16 + l];
            acc[t] = __builtin_amdgcn_wmma_f32_16x16x4_f32(
                /*neg_a=*/false, a, /*neg_b=*/false, b,
                /*c_mod=*/(short)0, acc[t], /*reuse_a=*/false, /*reuse_b=*/false);
        }
    }

#pragma unroll
    for (int r = 0; r < 8; ++r) {
        const int row = m0 + r + half * 8;
        const float s = dinv[row];
        float* __restrict__ orow = G + (size_t)row * NCOL + l;
#pragma unroll
        for (int t = 0; t < NT; ++t) orow[t * 16] = acc[t][r] * s;
    }
}

// ---------------- zero fill ----------------
__global__ void k_zero(float* __restrict__ p, int n) {
    int i = blockIdx.x * blockDim.x + threadIdx.x;
    if (i < n) p[i] = 0.0f;
}

// ---------------- edge scatter: acc[dst] += g[src] (float4 per thread) ----------------
template <int NCOL>
__global__ void k_scatter(const long long* __restrict__ src,
                          const long long* __restrict__ dst,
                          const float* __restrict__ G,
                          float* __restrict__ acc) {
    constexpr int TPE = NCOL / 4;            // threads per edge
    const int gid = blockIdx.x * blockDim.x + threadIdx.x;
    const int e = gid / TPE;
    const int f = (gid % TPE) * 4;
    if (e >= NEDGES) return;
    const int s = (int)src[e];
    const int d = (int)dst[e];
    const float4 v = *(const float4*)(G + (size_t)s * NCOL + f);
    float* __restrict__ o = acc + (size_t)d * NCOL + f;
    atomicAdd(o + 0, v.x);
    atomicAdd(o + 1, v.y);
    atomicAdd(o + 2, v.z);
    atomicAdd(o + 3, v.w);
}

// ---------------- epilogue: out = dinv[row]*(acc + g_selfloop) + bias (optional ReLU) ----
template <int NCOL, bool RELU>
__global__ void k_epilogue(const float* __restrict__ acc,
                           const float* __restrict__ G,
                           const float* __restrict__ dinv,
                           const float* __restrict__ bias,
                           float* __restrict__ out) {
    const int i = blockIdx.x * blockDim.x + threadIdx.x;
    if (i >= NNODES * NCOL) return;
    const int row = i / NCOL;
    const int col = i - row * NCOL;
    float v = dinv[row] * (acc[i] + G[i]) + bias[col];
    if (RELU) v = fmaxf(v, 0.0f);
    out[i] = v;
}

// ---------------- host launch ----------------
extern "C" void kernel_launch(void* const* d_in, const int* in_sizes, int n_in,
                              void* d_out, int out_size, void* d_ws, size_t ws_size,
                              hipStream_t stream) {
    const float*     x   = (const float*)d_in[0];
    const long long* ei  = (const long long*)d_in[1];   // [2, E] int64; row0=src row1=dst
    const float*     W1  = (const float*)d_in[2];
    const float*     b1  = (const float*)d_in[3];
    const float*     W2  = (const float*)d_in[4];
    const float*     b2  = (const float*)d_in[5];
    float*           out = (float*)d_out;

    const long long* esrc = ei;
    const long long* edst = ei + NEDGES;

    // workspace layout (floats): dinv | g1[N*128] | accbuf[N*128] (layer2: g2|acc2 halves)
    float* ws   = (float*)d_ws;
    float* dinv = ws;
    float* g1   = ws + 50048;                         // 64-aligned past 50000
    float* acc1 = g1 + (size_t)NNODES * C_HID;
    float* g2   = acc1;                               // reuse after acc1 consumed
    float* acc2 = acc1 + (size_t)NNODES * C_OUT;

    const int T = 256;

    // degree -> dinv (deg includes self-loop)
    k_init_deg<<<(NNODES + T - 1) / T, T, 0, stream>>>(dinv);
    k_deg_edges<<<(NEDGES + T - 1) / T, T, 0, stream>>>(edst, dinv);
    k_dinv<<<(NNODES + T - 1) / T, T, 0, stream>>>(dinv);

    const int waves  = NNODES / 16;                   // 3125 row strips
    const int gblk   = (waves * 32 + T - 1) / T;      // 8 waves / block

    // ---- layer 1: g1 = (x @ W1) * dinv ----
    k_gemm_scale<C_IN, C_HID><<<gblk, T, 0, stream>>>(x, W1, dinv, g1);
    k_zero<<<((NNODES * C_HID) + T - 1) / T, T, 0, stream>>>(acc1, NNODES * C_HID);
    {
        const long long nthr = (long long)NEDGES * (C_HID / 4);
        k_scatter<C_HID><<<(int)((nthr + T - 1) / T), T, 0, stream>>>(esrc, edst, g1, acc1);
    }
    // h1 (ReLU) written in-place over g1
    k_epilogue<C_HID, true><<<((NNODES * C_HID) + T - 1) / T, T, 0, stream>>>(acc1, g1, dinv, b1, g1);

    // ---- layer 2: g2 = (h1 @ W2) * dinv ----
    k_gemm_scale<C_HID, C_OUT><<<gblk, T, 0, stream>>>(g1, W2, dinv, g2);
    k_zero<<<((NNODES * C_OUT) + T - 1) / T, T, 0, stream>>>(acc2, NNODES * C_OUT);
    {
        const long long nthr = (long long)NEDGES * (C_OUT / 4);
        k_scatter<C_OUT><<<(int)((nthr + T - 1) / T), T, 0, stream>>>(esrc, edst, g2, acc2);
    }
    k_epilogue<C_OUT, false><<<((NNODES * C_OUT) + T - 1) / T, T, 0, stream>>>(acc2, g2, dinv, b2, out);
}